// GraphModel_16801912062662
// MI455X (gfx1250) — compile-verified
//
#include <hip/hip_runtime.h>
#include <hip/hip_bf16.h>
#include <stddef.h>

// ---------------------------------------------------------------------------
// GCNConv (D^-1/2 (A+I) D^-1/2 X W + b)[x]  for MI455X (gfx1250, wave32).
//
// Pipeline (all on `stream`, graph-capture safe):
//   K0 init     : out_full = 0, deg = 1 (self loop), mask = 0
//   K1 mark     : mask[x[i]] = 1            (only ~15% of nodes are ever read)
//   K2 degree   : deg[dst] += 1 over 3.2M edges (f32 atomics)
//   K3 rsqrt    : deg -> dinv in place
//   K4 gemm     : h = X @ W via V_WMMA_F32_16X16X4_F32, one wave per 16x16 tile
//   K5 message  : for masked dst only: out_full[dst] += dinv[src]*dinv[dst]*h[src]
//   K6 gather   : out[i] = out_full[node] + dinv[node]^2 * h[node] + bias
//
// Workspace (~52 MB): deg[100000] f32 | mask[100000] u32 | h[100000x64] f32 |
//                     out_full[100000x64] f32
// ---------------------------------------------------------------------------

#define N_NODES   100000
#define FIELD_DIM 128
#define EMBED_DIM 64
#define N_EDGES   3200000
#define BATCH     8192
#define NUM_FIELDS 2

typedef __attribute__((ext_vector_type(2))) float v2f;
typedef __attribute__((ext_vector_type(8))) float v8f;

static __device__ __forceinline__ void atomic_add_f32(float* p, float v) {
    __hip_atomic_fetch_add(p, v, __ATOMIC_RELAXED, __HIP_MEMORY_SCOPE_AGENT);
}

// K0: zero the accumulator, seed degree with the self loop, clear the mask.
__global__ void __launch_bounds__(256) k_init(float* __restrict__ out_full,
                                              float* __restrict__ deg,
                                              unsigned* __restrict__ mask) {
    const int i = blockIdx.x * 256 + threadIdx.x;
    if (i < N_NODES * EMBED_DIM) out_full[i] = 0.0f;
    if (i < N_NODES) { deg[i] = 1.0f; mask[i] = 0u; }
}

// K1: mark destination nodes that the final gather will actually read.
__global__ void __launch_bounds__(256) k_mark(const int* __restrict__ x,
                                              unsigned* __restrict__ mask) {
    const int i = blockIdx.x * 256 + threadIdx.x;
    if (i < BATCH * NUM_FIELDS) mask[x[i]] = 1u;
}

// K2: degree over all edges (dst row of edge_index).
__global__ void __launch_bounds__(256) k_degree(const int* __restrict__ ei,
                                                float* __restrict__ deg) {
    const int e = blockIdx.x * 256 + threadIdx.x;
    if (e < N_EDGES) atomic_add_f32(&deg[ei[N_EDGES + e]], 1.0f);
}

// K3: deg -> rsqrt(deg) in place (deg >= 1, safe).
__global__ void __launch_bounds__(256) k_rsqrt(float* __restrict__ deg) {
    const int i = blockIdx.x * 256 + threadIdx.x;
    if (i < N_NODES) deg[i] = rsqrtf(deg[i]);
}

// K4: h = X @ W with V_WMMA_F32_16X16X4_F32. One wave32 computes one 16x16
// tile of h; K loop of 32 WMMAs covers FIELD_DIM=128.
//   A (16x4): lanes 0-15 hold M=0..15 with K={0,1} in the 2 VGPRs,
//             lanes 16-31 hold K={2,3}.
//   B (4x16): lanes 0-15 hold N=0..15 with K={0,1}, lanes 16-31 K={2,3}.
//   C/D     : 8 VGPRs; lane<16 -> rows 0..7, lane>=16 -> rows 8..15.
__global__ void __launch_bounds__(256) k_gemm(const float* __restrict__ X,
                                              const float* __restrict__ W,
                                              float* __restrict__ H) {
    const int lane  = threadIdx.x & 31;
    const int wave  = threadIdx.x >> 5;
    const int wid   = blockIdx.x * 8 + wave;
    const int NT    = EMBED_DIM / 16;             // 4 tiles across N
    const int MT    = N_NODES / 16;               // 6250 tiles down M
    if (wid >= MT * NT) return;
    const int tm = wid / NT;
    const int tn = wid - tm * NT;

    const int l15   = lane & 15;
    const int khalf = (lane >> 4) * 2;            // 0 for lanes 0-15, 2 for 16-31
    const float* xrow = X + (size_t)(tm * 16 + l15) * FIELD_DIM + khalf;
    const float* wcol = W + (size_t)khalf * EMBED_DIM + (tn * 16 + l15);

    v8f acc = {};
    #pragma unroll 4
    for (int k = 0; k < FIELD_DIM; k += 4) {
        v2f a, b;
        a.x = xrow[k + 0];
        a.y = xrow[k + 1];
        b.x = wcol[(size_t)k * EMBED_DIM];
        b.y = wcol[(size_t)(k + 1) * EMBED_DIM];
        acc = __builtin_amdgcn_wmma_f32_16x16x4_f32(
            /*neg_a=*/false, a, /*neg_b=*/false, b,
            /*c_mod=*/(short)0, acc, /*reuse_a=*/false, /*reuse_b=*/false);
    }

    // Store the 16x16 f32 D tile per the ISA C/D layout.
    const int srow = tm * 16 + (lane >> 4) * 8;
    const int scol = tn * 16 + l15;
    float* hout = H + (size_t)srow * EMBED_DIM + scol;
    #pragma unroll
    for (int r = 0; r < 8; ++r)
        hout[(size_t)r * EMBED_DIM] = acc[r];
}

// K5: edge aggregation, 16 threads per edge, 4 columns each (float4 gather,
// 4 f32 atomics). Edges whose destination is never gathered are skipped,
// cutting the dominant HBM traffic ~6.5x.
__global__ void __launch_bounds__(256) k_message(const int* __restrict__ ei,
                                                 const float* __restrict__ dinv,
                                                 const unsigned* __restrict__ mask,
                                                 const float* __restrict__ H,
                                                 float* __restrict__ out_full) {
    const int t = blockIdx.x * 256 + threadIdx.x;
    const int e = t >> 4;
    if (e >= N_EDGES) return;
    const int dst = ei[N_EDGES + e];
    if (!mask[dst]) return;
    const int src  = ei[e];
    const float nr = dinv[src] * dinv[dst];
    const int part = (t & 15) * 4;
    const float4 hv = *(const float4*)(H + (size_t)src * EMBED_DIM + part);
    float* o = out_full + (size_t)dst * EMBED_DIM + part;
    atomic_add_f32(o + 0, hv.x * nr);
    atomic_add_f32(o + 1, hv.y * nr);
    atomic_add_f32(o + 2, hv.z * nr);
    atomic_add_f32(o + 3, hv.w * nr);
}

// K6: final gather, folding in the self-loop term dinv^2 * h and the bias.
__global__ void __launch_bounds__(256) k_gather(const int* __restrict__ x,
                                                const float* __restrict__ out_full,
                                                const float* __restrict__ H,
                                                const float* __restrict__ dinv,
                                                const float* __restrict__ bias,
                                                float* __restrict__ out) {
    const int i = blockIdx.x * 256 + threadIdx.x;
    if (i >= BATCH * NUM_FIELDS * EMBED_DIM) return;
    const int c    = i & (EMBED_DIM - 1);
    const int bf   = i >> 6;                       // EMBED_DIM == 64
    const int node = x[bf];
    const float di = dinv[node];
    const size_t off = (size_t)node * EMBED_DIM + c;
    out[i] = out_full[off] + di * di * H[off] + bias[c];
}

extern "C" void kernel_launch(void* const* d_in, const int* in_sizes, int n_in,
                              void* d_out, int out_size, void* d_ws, size_t ws_size,
                              hipStream_t stream) {
    const float* features = (const float*)d_in[0];   // [N_NODES, FIELD_DIM]
    const int*   ei       = (const int*)d_in[1];     // [2, N_EDGES]
    const float* W        = (const float*)d_in[2];   // [FIELD_DIM, EMBED_DIM]
    const float* bias     = (const float*)d_in[3];   // [EMBED_DIM]
    const int*   x        = (const int*)d_in[4];     // [BATCH, NUM_FIELDS]
    float*       out      = (float*)d_out;           // [BATCH, NUM_FIELDS, EMBED_DIM]

    // Workspace layout (all offsets 256B aligned).
    char* ws = (char*)d_ws;
    float*    deg      = (float*)(ws);                         // 400 KB (becomes dinv)
    unsigned* mask     = (unsigned*)(ws + 409600);             // 400 KB
    float*    h        = (float*)(ws + 819200);                // 25.6 MB
    float*    out_full = (float*)(ws + 819200 + 25600000);     // 25.6 MB
    (void)in_sizes; (void)n_in; (void)out_size; (void)ws_size;

    // K0 init: covers N_NODES*EMBED_DIM elements.
    k_init<<<(N_NODES * EMBED_DIM + 255) / 256, 256, 0, stream>>>(out_full, deg, mask);
    // K1 mark gathered destinations.
    k_mark<<<(BATCH * NUM_FIELDS + 255) / 256, 256, 0, stream>>>(x, mask);
    // K2 degree.
    k_degree<<<(N_EDGES + 255) / 256, 256, 0, stream>>>(ei, deg);
    // K3 dinv.
    k_rsqrt<<<(N_NODES + 255) / 256, 256, 0, stream>>>(deg);
    // K4 GEMM: 6250*4 = 25000 wave-tiles, 8 waves per 256-thread block.
    k_gemm<<<(25000 + 7) / 8, 256, 0, stream>>>(features, W, h);
    // K5 aggregation: 16 threads per edge.
    {
        const long long threads = (long long)N_EDGES * 16;
        k_message<<<(unsigned)((threads + 255) / 256), 256, 0, stream>>>(ei, deg, mask, h, out_full);
    }
    // K6 gather.
    k_gather<<<(BATCH * NUM_FIELDS * EMBED_DIM + 255) / 256, 256, 0, stream>>>(
        x, out_full, h, deg, bias, out);
}